// GAT_34883724378268
// MI455X (gfx1250) — compile-verified
//
#include <hip/hip_runtime.h>
#include <hip/hip_bf16.h>
#include <math.h>

#define Nn 50000
#define Ee 800000
#define Gg 64

typedef __attribute__((ext_vector_type(2))) float v2f;
typedef __attribute__((ext_vector_type(8))) float v8f;

// ---------------- monotonic float<->uint encoding for atomic max ----------------
__device__ __forceinline__ unsigned fenc(float f) {
    unsigned u = __float_as_uint(f);
    return (u & 0x80000000u) ? ~u : (u | 0x80000000u);
}
__device__ __forceinline__ float fdec(unsigned u) {
    u = (u & 0x80000000u) ? (u & 0x7FFFFFFFu) : ~u;
    return __uint_as_float(u);
}

// ---------------- zero fill ----------------
__global__ void fill_zero_k(float* p, long long n) {
    long long i = (long long)blockIdx.x * blockDim.x + threadIdx.x;
    if (i < n) p[i] = 0.0f;
}

// ---------------- fp32 WMMA GEMM: out[M x NC] = A[M x K] @ W[K x NC] + bias ----
// 16 rows per block; each wave owns 16-col tiles; V_WMMA_F32_16X16X4_F32 chain.
template <int K, int NC>
__global__ void gemm_bias_wmma(const float* __restrict__ A,
                               const float* __restrict__ W,
                               const float* __restrict__ bias,
                               float* __restrict__ out) {
    constexpr int LDA = K + 4;               // conflict-free: (lm*LDA)%64 = 4*lm%64
    __shared__ float As[16 * LDA];
    const int tid  = threadIdx.x;
    const int row0 = blockIdx.x * 16;

    for (int i = tid; i < 16 * K; i += blockDim.x) {
        int r = i / K, c = i - r * K;
        As[r * LDA + c] = A[(size_t)(row0 + r) * K + c];
    }
    __syncthreads();

    const int wave = tid >> 5, lane = tid & 31, nwaves = blockDim.x >> 5;
    const int half = lane >> 4, lm = lane & 15;

    for (int ct = wave; ct < NC / 16; ct += nwaves) {
        const int col = ct * 16 + lm;
        v8f acc = {};
        const float* as = &As[lm * LDA + 2 * half];          // A frag: lanes<16 K=0,1 ; lanes>=16 K=2,3
        const float* wp = &W[(size_t)(2 * half) * NC + col]; // B frag: same K split, N=lm
#pragma unroll
        for (int k0 = 0; k0 < K; k0 += 4) {
            v2f a, b;
            a[0] = as[k0];
            a[1] = as[k0 + 1];
            b[0] = wp[(size_t)k0 * NC];
            b[1] = wp[(size_t)(k0 + 1) * NC];
            acc = __builtin_amdgcn_wmma_f32_16x16x4_f32(false, a, false, b,
                                                        (short)0, acc, false, false);
        }
        const float bc = bias[col];
#pragma unroll
        for (int r = 0; r < 8; ++r)
            out[(size_t)(row0 + half * 8 + r) * NC + col] = acc[r] + bc;
    }
}

// ---------------- layer-1 edge scores (H=8, C=32) + segment max ----------------
__global__ void edge_score8_k(const float* __restrict__ xl, const float* __restrict__ xr,
                              const float* __restrict__ We, const float* __restrict__ att,
                              const float* __restrict__ ew, const int* __restrict__ ei,
                              float* __restrict__ s, unsigned* __restrict__ menc) {
    int idx = blockIdx.x * blockDim.x + threadIdx.x;
    if (idx >= Ee * 8) return;
    int e = idx >> 3, h = idx & 7;
    int src = ei[e], dst = ei[Ee + e];
    float w = ew[e];
    const float* pl = xl + (size_t)src * 256 + h * 32;
    const float* pr = xr + (size_t)dst * 256 + h * 32;
    const float* pw = We + h * 32;
    const float* pa = att + h * 32;
    float sc = 0.0f;
#pragma unroll
    for (int c = 0; c < 32; ++c) {
        float v = pl[c] + pr[c] + w * pw[c];
        v = v > 0.0f ? v : 0.2f * v;                   // leaky_relu(0.2)
        sc += v * pa[c];
    }
    s[idx] = sc;
    atomicMax(&menc[(size_t)dst * 8 + h], fenc(sc));
}

// ---------------- layer-2 edge scores (H=1, C=32) ----------------
__global__ void edge_score1_k(const float* __restrict__ xl, const float* __restrict__ xr,
                              const float* __restrict__ We, const float* __restrict__ att,
                              const float* __restrict__ ew, const int* __restrict__ ei,
                              float* __restrict__ s, unsigned* __restrict__ menc) {
    int e = blockIdx.x * blockDim.x + threadIdx.x;
    if (e >= Ee) return;
    int src = ei[e], dst = ei[Ee + e];
    float w = ew[e];
    const float* pl = xl + (size_t)src * 32;
    const float* pr = xr + (size_t)dst * 32;
    float sc = 0.0f;
#pragma unroll
    for (int c = 0; c < 32; ++c) {
        float v = pl[c] + pr[c] + w * We[c];
        v = v > 0.0f ? v : 0.2f * v;
        sc += v * att[c];
    }
    s[e] = sc;
    atomicMax(&menc[dst], fenc(sc));
}

// ---------------- decode segment max, non-finite -> 0 (in place) ----------------
__global__ void node_max_fix_k(unsigned* menc, int n) {
    int i = blockIdx.x * blockDim.x + threadIdx.x;
    if (i >= n) return;
    float m = fdec(menc[i]);
    if (!isfinite(m)) m = 0.0f;
    ((float*)menc)[i] = m;
}

// ---------------- p = exp(s - m[dst]), denom += p ----------------
template <int H>
__global__ void edge_exp_k(const float* __restrict__ m, const int* __restrict__ ei,
                           float* __restrict__ s, float* __restrict__ denom) {
    int idx = blockIdx.x * blockDim.x + threadIdx.x;
    if (idx >= Ee * H) return;
    int e = idx / H, h = idx - e * H;
    int dst = ei[Ee + e];
    float p = expf(s[idx] - m[(size_t)dst * H + h]);
    s[idx] = p;
    atomicAdd(&denom[(size_t)dst * H + h], p);
}

// ---------------- layer-1 aggregate: one block (256 lanes) per edge ----------------
__global__ void edge_agg8_k(const float* __restrict__ xl, const float* __restrict__ p,
                            const float* __restrict__ denom, const int* __restrict__ ei,
                            float* __restrict__ out) {
    int e = blockIdx.x;
    int c = threadIdx.x;                      // 0..255
    int h = c >> 5;
    int src = ei[e], dst = ei[Ee + e];
    float alpha = p[(size_t)e * 8 + h] / denom[(size_t)dst * 8 + h];
    atomicAdd(&out[(size_t)dst * 256 + c], alpha * xl[(size_t)src * 256 + c]);
}

// ---------------- layer-2 aggregate: 32 lanes per edge ----------------
__global__ void edge_agg1_k(const float* __restrict__ xl, const float* __restrict__ p,
                            const float* __restrict__ denom, const int* __restrict__ ei,
                            float* __restrict__ out) {
    int idx = blockIdx.x * blockDim.x + threadIdx.x;
    int e = idx >> 5, c = idx & 31;
    if (e >= Ee) return;
    int src = ei[e], dst = ei[Ee + e];
    float alpha = p[e] / denom[dst];
    atomicAdd(&out[(size_t)dst * 32 + c], alpha * xl[(size_t)src * 32 + c]);
}

// ---------------- bias add (+ optional ELU) ----------------
template <int CMASK, int ELU>
__global__ void bias_act_k(float* __restrict__ h, const float* __restrict__ bias, int total) {
    int idx = blockIdx.x * blockDim.x + threadIdx.x;
    if (idx >= total) return;
    float v = h[idx] + bias[idx & CMASK];
    if (ELU) v = v > 0.0f ? v : expf(v) - 1.0f;
    h[idx] = v;
}

// ---------------- mean-pool accumulation ----------------
__global__ void pool_k(const float* __restrict__ h2, const int* __restrict__ batch,
                       float* __restrict__ sums, float* __restrict__ cnt) {
    int idx = blockIdx.x * blockDim.x + threadIdx.x;
    if (idx >= Nn * 32) return;
    int n = idx >> 5, c = idx & 31;
    int g = batch[n];
    atomicAdd(&sums[(size_t)g * 32 + c], h2[idx]);
    if (c == 0) atomicAdd(&cnt[g], 1.0f);
}

// ---------------- head: mean, linear(32->10), log_softmax; 1 block, 64 lanes ----
__global__ void head_k(const float* __restrict__ sums, const float* __restrict__ cnt,
                       const float* __restrict__ Wlin, const float* __restrict__ blin,
                       float* __restrict__ out) {
    int g = threadIdx.x;
    if (g >= Gg) return;
    float inv = 1.0f / fmaxf(cnt[g], 1.0f);
    float lg[10];
    float mx = -3.0e38f;
#pragma unroll
    for (int j = 0; j < 10; ++j) {
        float a = blin[j];
#pragma unroll
        for (int c = 0; c < 32; ++c) a += sums[(size_t)g * 32 + c] * inv * Wlin[c * 10 + j];
        lg[j] = a;
        mx = fmaxf(mx, a);
    }
    float se = 0.0f;
#pragma unroll
    for (int j = 0; j < 10; ++j) se += expf(lg[j] - mx);
    float ls = logf(se);
#pragma unroll
    for (int j = 0; j < 10; ++j) out[g * 10 + j] = lg[j] - mx - ls;
}

static inline int cdiv(long long a, int b) { return (int)((a + b - 1) / b); }

extern "C" void kernel_launch(void* const* d_in, const int* in_sizes, int n_in,
                              void* d_out, int out_size, void* d_ws, size_t ws_size,
                              hipStream_t stream) {
    (void)in_sizes; (void)n_in; (void)out_size; (void)ws_size;
    const float* x     = (const float*)d_in[0];
    const float* ew    = (const float*)d_in[1];
    const float* Wl1   = (const float*)d_in[2];
    const float* bl1   = (const float*)d_in[3];
    const float* Wr1   = (const float*)d_in[4];
    const float* br1   = (const float*)d_in[5];
    const float* We1   = (const float*)d_in[6];
    const float* att1  = (const float*)d_in[7];
    const float* bias1 = (const float*)d_in[8];
    const float* Wl2   = (const float*)d_in[9];
    const float* bl2   = (const float*)d_in[10];
    const float* Wr2   = (const float*)d_in[11];
    const float* br2   = (const float*)d_in[12];
    const float* We2   = (const float*)d_in[13];
    const float* att2  = (const float*)d_in[14];
    const float* bias2 = (const float*)d_in[15];
    const float* Wlin  = (const float*)d_in[16];
    const float* blin  = (const float*)d_in[17];
    const int*   ei    = (const int*)d_in[18];
    const int*   batch = (const int*)d_in[19];
    float* out = (float*)d_out;

    // workspace layout (floats)
    float* ws = (float*)d_ws;
    const size_t NHC = (size_t)Nn * 256;
    float*    xl1 = ws;                                    // N*256
    float*    xr1 = ws + NHC;                              // N*256
    float*    h1  = ws + 2 * NHC;                          // N*256 (layer1 accumulator)
    float*    s1  = ws + 3 * NHC;                          // E*8
    unsigned* m1  = (unsigned*)(ws + 3 * NHC + (size_t)Ee * 8);   // N*8
    float*    d1  = ws + 3 * NHC + (size_t)Ee * 8 + (size_t)Nn * 8; // N*8
    // layer-2 reuse (layer-1 regions are dead by then)
    float*    xl2 = ws;                                    // N*32
    float*    xr2 = ws + (size_t)Nn * 32;                  // N*32
    float*    h2  = ws + (size_t)Nn * 64;                  // N*32
    float*    s2  = s1;                                    // E
    unsigned* m2  = m1;                                    // N
    float*    d2  = d1;                                    // N
    float*    sums = xr1;                                  // G*32
    float*    cnt  = xr1 + Gg * 32;                        // G

    const int B = 256;
    const int rowTiles = Nn / 16;   // 50000 / 16 = 3125 exactly

    // ---- layer 1 projections (fp32 WMMA) ----
    gemm_bias_wmma<128, 256><<<rowTiles, 256, 0, stream>>>(x, Wl1, bl1, xl1);
    gemm_bias_wmma<128, 256><<<rowTiles, 256, 0, stream>>>(x, Wr1, br1, xr1);

    // ---- zero max/denom/accumulator ----
    fill_zero_k<<<cdiv((long long)Nn * 16, B), B, 0, stream>>>((float*)m1, (long long)Nn * 16);
    fill_zero_k<<<cdiv((long long)NHC, B), B, 0, stream>>>(h1, (long long)NHC);

    // ---- layer 1 softmax-attention aggregation ----
    edge_score8_k<<<cdiv((long long)Ee * 8, B), B, 0, stream>>>(xl1, xr1, We1, att1, ew, ei, s1, m1);
    node_max_fix_k<<<cdiv((long long)Nn * 8, B), B, 0, stream>>>(m1, Nn * 8);
    edge_exp_k<8><<<cdiv((long long)Ee * 8, B), B, 0, stream>>>((const float*)m1, ei, s1, d1);
    edge_agg8_k<<<Ee, 256, 0, stream>>>(xl1, s1, d1, ei, h1);
    bias_act_k<255, 1><<<cdiv((long long)NHC, B), B, 0, stream>>>(h1, bias1, (int)NHC);

    // ---- layer 2 projections (fp32 WMMA, K=256, NC=32) ----
    gemm_bias_wmma<256, 32><<<rowTiles, 64, 0, stream>>>(h1, Wl2, bl2, xl2);
    gemm_bias_wmma<256, 32><<<rowTiles, 64, 0, stream>>>(h1, Wr2, br2, xr2);

    fill_zero_k<<<cdiv((long long)Nn * 16, B), B, 0, stream>>>((float*)m2, (long long)Nn * 16);
    fill_zero_k<<<cdiv((long long)Nn * 32, B), B, 0, stream>>>(h2, (long long)Nn * 32);

    // ---- layer 2 softmax-attention aggregation ----
    edge_score1_k<<<cdiv(Ee, B), B, 0, stream>>>(xl2, xr2, We2, att2, ew, ei, s2, m2);
    node_max_fix_k<<<cdiv(Nn, B), B, 0, stream>>>(m2, Nn);
    edge_exp_k<1><<<cdiv(Ee, B), B, 0, stream>>>((const float*)m2, ei, s2, d2);
    edge_agg1_k<<<cdiv((long long)Ee * 32, B), B, 0, stream>>>(xl2, s2, d2, ei, h2);
    bias_act_k<31, 0><<<cdiv((long long)Nn * 32, B), B, 0, stream>>>(h2, bias2, Nn * 32);

    // ---- mean pool + linear + log_softmax ----
    fill_zero_k<<<cdiv(Gg * 33, B), B, 0, stream>>>(sums, Gg * 33);
    pool_k<<<cdiv((long long)Nn * 32, B), B, 0, stream>>>(h2, batch, sums, cnt);
    head_k<<<1, 64, 0, stream>>>(sums, cnt, Wlin, blin, out);
}